// DOnePole_37641093382427
// MI455X (gfx1250) — compile-verified
//
#include <hip/hip_runtime.h>

typedef __attribute__((ext_vector_type(2))) float v2f;
typedef __attribute__((ext_vector_type(8))) float v8f;

#define CH     256     // batch channels (fixed by reference)
#define LCHUNK 2048    // timesteps per chunk (one wave-task)
#define TILE   256     // timesteps per WMMA tile (16 segments x 16 steps)

__device__ __forceinline__ float powa(float a, int e) {
  float r = 1.0f;
  for (int k = 0; k < e; ++k) r *= a;
  return r;
}

__device__ __forceinline__ float clamp_a1(float a) {
  return fminf(fmaxf(a, -1.0f), 1.0f);
}

// ---------------- Pass 1: per-chunk zero-state contribution delta ------------
__global__ void onepole_pass1(const float* __restrict__ x,
                              const float* __restrict__ b0p,
                              const float* __restrict__ b1p,
                              const float* __restrict__ a1p,
                              float* __restrict__ delta,
                              int T, int G) {
  const int lane = threadIdx.x & 31;
  const int wave = blockIdx.x * (blockDim.x >> 5) + (threadIdx.x >> 5);
  const int ch = wave / G;
  const int g  = wave % G;
  const long long base = (long long)ch * T + (long long)g * LCHUNK;

  const float a1 = clamp_a1(a1p[0]);
  const float c  = b1p[0] + a1 * b0p[0];
  const float a8   = powa(a1, 8);
  const float a256 = powa(a8, 32);

  // lane covers tile-local elements lane*8 .. lane*8+7 ; weight = a1^(255 - i)
  float w[8];
  w[7] = powa(a8, 31 - lane);          // a1^(248 - 8*lane)
  for (int j = 6; j >= 0; --j) w[j] = w[j + 1] * a1;

  float acc = 0.0f;
  const int ntile = LCHUNK / TILE;
  for (int t = 0; t < ntile; ++t) {
    const float* p = x + base + t * TILE + lane * 8;
    if (t + 1 < ntile) __builtin_prefetch(p + TILE, 0, 1);
    float4 u = *(const float4*)p;
    float4 v = *(const float4*)(p + 4);
    float partial =            w[0] * u.x;
    partial = fmaf(w[1], u.y, partial);
    partial = fmaf(w[2], u.z, partial);
    partial = fmaf(w[3], u.w, partial);
    partial = fmaf(w[4], v.x, partial);
    partial = fmaf(w[5], v.y, partial);
    partial = fmaf(w[6], v.z, partial);
    partial = fmaf(w[7], v.w, partial);
    acc = fmaf(a256, acc, partial);    // fold tile into running chunk sum
  }
  // wave reduction (wave32)
  for (int off = 16; off; off >>= 1) acc += __shfl_xor(acc, off, 32);
  if (lane == 0) delta[ch * G + g] = c * acc;
}

// ---------------- Pass 2: serial chain of chunk states per channel ----------
__global__ void onepole_pass2(const float* __restrict__ delta,
                              float* __restrict__ sstart,
                              const float* __restrict__ a1p,
                              int G) {
  const int ch = threadIdx.x;           // 256 threads, one per channel
  const float a1 = clamp_a1(a1p[0]);
  const float a256 = powa(powa(a1, 16), 16);
  const float AL = powa(a256, LCHUNK / 256);   // a1^LCHUNK
  float s = 0.0f;
  for (int g = 0; g < G; ++g) {
    const int i = ch * G + g;
    sstart[i] = s;
    s = fmaf(AL, s, delta[i]);
  }
}

// ---------------- Pass 3: WMMA tile evaluation + segment scan + store -------
__global__ void onepole_pass3(const float* __restrict__ x,
                              const float* __restrict__ b0p,
                              const float* __restrict__ b1p,
                              const float* __restrict__ a1p,
                              const float* __restrict__ sstart,
                              float* __restrict__ y,
                              int T, int G) {
  __shared__ float lds[4 * TILE];       // 4 waves per 128-thread block

  const int lane = threadIdx.x & 31;
  const int wave = blockIdx.x * (blockDim.x >> 5) + (threadIdx.x >> 5);
  const int ch = wave / G;
  const int g  = wave % G;
  const long long base = (long long)ch * T + (long long)g * LCHUNK;

  const int seg = lane & 15;            // WMMA column / A row for this lane
  const int hi  = (lane >> 4) & 1;

  const float a1  = clamp_a1(a1p[0]);
  const float b0v = b0p[0];
  const float b1v = b1p[0];
  const float c   = b1v + a1 * b0v;

  // A-matrix slices: M[i][j] = (j==i)?b0 : (j<i ? c*a1^(i-1-j) : 0); row i = seg
  v2f aA[4];
#pragma unroll
  for (int q = 0; q < 4; ++q) {
    const int k0 = 4 * q + 2 * hi;
#pragma unroll
    for (int e = 0; e < 2; ++e) {
      const int j = k0 + e;
      float m;
      if (j == seg)      m = b0v;
      else if (j < seg)  m = c * powa(a1, seg - 1 - j);
      else               m = 0.0f;
      if (e == 0) aA[q].x = m; else aA[q].y = m;
    }
  }

  // fix-up / scan constants
  const float m1 = powa(a1, 16);
  const float m2 = m1 * m1;
  const float m4 = m2 * m2;
  const float m8 = m4 * m4;
  const float a256 = m8 * m8;
  const float p16seg = powa(m1, seg);   // a1^(16*seg)
  float rowpow[8];
#pragma unroll
  for (int v = 0; v < 8; ++v) rowpow[v] = powa(a1, v + 8 * hi);

  float s_in = sstart[ch * G + g];      // state entering this chunk
  float* lb = lds + (threadIdx.x >> 5) * TILE;

  const int ntile = LCHUNK / TILE;
  for (int t = 0; t < ntile; ++t) {
    const float* xt = x + base + t * TILE;
    if (t + 1 < ntile) __builtin_prefetch(xt + TILE + lane * 8, 0, 1);

    // B-matrix: lane holds column 'seg', K = 4q + 2*hi + {0,1}
    v2f bq[4];
#pragma unroll
    for (int q = 0; q < 4; ++q) {
      const int idx = seg * 16 + 4 * q + 2 * hi;
      bq[q].x = xt[idx];
      bq[q].y = xt[idx + 1];
    }

    // Y = M * X via 4 chained 16x16x4 f32 WMMAs
    v8f cacc = {};
#pragma unroll
    for (int q = 0; q < 4; ++q) {
      cacc = __builtin_amdgcn_wmma_f32_16x16x4_f32(
          false, aA[q], false, bq[q], (short)0, cacc, false, false);
    }

    // per-segment zero-state end state: delta_seg = b1*x15 + a1*Y[15][seg]
    // (valid on hi lanes: they hold row M=15 in cacc[7] and x15 in bq[3].y)
    float d = fmaf(b1v, bq[3].y, a1 * cacc[7]);
    float S = __shfl(d, 16 + seg, 32);  // everyone picks up its segment's delta

    // Kogge-Stone inclusive scan over 16 segments (replicated in both halves)
    float u;
    u = __shfl(S, (lane & 16) | ((seg - 1) & 15), 32); S = (seg >= 1) ? fmaf(m1, u, S) : S;
    u = __shfl(S, (lane & 16) | ((seg - 2) & 15), 32); S = (seg >= 2) ? fmaf(m2, u, S) : S;
    u = __shfl(S, (lane & 16) | ((seg - 4) & 15), 32); S = (seg >= 4) ? fmaf(m4, u, S) : S;
    u = __shfl(S, (lane & 16) | ((seg - 8) & 15), 32); S = (seg >= 8) ? fmaf(m8, u, S) : S;

    const float Sprev = __shfl(S, (lane & 16) | ((seg - 1) & 15), 32);
    const float s_seg = fmaf(p16seg, s_in, (seg == 0) ? 0.0f : Sprev);
    const float S15   = __shfl(S, 15, 32);

    // apply fix-up out[i][seg] = Y + a1^i * s_seg ; transpose through LDS
    float4 w0, w1;
    w0.x = fmaf(rowpow[0], s_seg, cacc[0]);
    w0.y = fmaf(rowpow[1], s_seg, cacc[1]);
    w0.z = fmaf(rowpow[2], s_seg, cacc[2]);
    w0.w = fmaf(rowpow[3], s_seg, cacc[3]);
    w1.x = fmaf(rowpow[4], s_seg, cacc[4]);
    w1.y = fmaf(rowpow[5], s_seg, cacc[5]);
    w1.z = fmaf(rowpow[6], s_seg, cacc[6]);
    w1.w = fmaf(rowpow[7], s_seg, cacc[7]);
    *(float4*)(lb + seg * 16 + 8 * hi)     = w0;
    *(float4*)(lb + seg * 16 + 8 * hi + 4) = w1;

    // coalesced 2x b128 store per lane (wave-private LDS; DS ops are in-order)
    float4 o0 = *(float4*)(lb + lane * 8);
    float4 o1 = *(float4*)(lb + lane * 8 + 4);
    *(float4*)(y + base + t * TILE + lane * 8)     = o0;
    *(float4*)(y + base + t * TILE + lane * 8 + 4) = o1;

    // carry to next tile
    s_in = fmaf(a256, s_in, S15);
  }
}

// ---------------------------------------------------------------------------
extern "C" void kernel_launch(void* const* d_in, const int* in_sizes, int n_in,
                              void* d_out, int out_size, void* d_ws, size_t ws_size,
                              hipStream_t stream) {
  (void)n_in; (void)out_size; (void)ws_size;
  const float* x  = (const float*)d_in[0];
  const float* b0 = (const float*)d_in[1];
  const float* b1 = (const float*)d_in[2];
  const float* a1 = (const float*)d_in[3];
  float* y = (float*)d_out;

  const int T = in_sizes[0] / CH;       // 131072 for the reference shapes
  const int G = T / LCHUNK;             // chunks per channel (64)

  float* delta  = (float*)d_ws;         // CH*G floats
  float* sstart = delta + CH * G;       // CH*G floats

  const int waves  = CH * G;            // 16384 independent wave-tasks
  dim3 blk(128);                        // 4 waves / block
  dim3 grd(waves / 4);

  onepole_pass1<<<grd, blk, 0, stream>>>(x, b0, b1, a1, delta, T, G);
  onepole_pass2<<<1, CH, 0, stream>>>(delta, sstart, a1, G);
  onepole_pass3<<<grd, blk, 0, stream>>>(x, b0, b1, a1, sstart, y, T, G);
}